// CenterLoss2_80367428042770
// MI455X (gfx1250) — compile-verified
//
#include <hip/hip_runtime.h>

#define N_TOK 262144
#define DIM   128
#define NCLS  64
#define ROWS  64                  // feature rows per slab
#define THREADS 128               // 4 wave32s
#define NUM_WG 512
#define SLABS (N_TOK / ROWS)      // 4096

typedef __attribute__((ext_vector_type(2))) float v2f;
typedef __attribute__((ext_vector_type(8))) float v8f;

__global__ void cl2_init_ws(float* __restrict__ diffSum,
                            unsigned* __restrict__ counts,
                            float* __restrict__ loss) {
    int t = blockIdx.x * blockDim.x + threadIdx.x;
    if (t < NCLS * DIM) diffSum[t] = 0.f;
    if (t < NCLS) counts[t] = 0u;
    if (t == 0) *loss = 0.f;
}

__global__ __launch_bounds__(THREADS)
void cl2_main(const float* __restrict__ feature,   // [N, D]
              const int*   __restrict__ label,     // [N]
              const float* __restrict__ centers,   // [C, D]
              float* __restrict__ dist,            // [N, C] output
              float* __restrict__ gDiff,           // [C*D] ws accum
              unsigned* __restrict__ gCnt,         // [C]  ws accum
              float* __restrict__ gLoss)           // [1]  ws accum
{
    // B operand pre-paired: sB[kp][c] = { centers[c][2kp], centers[c][2kp+1] }
    // row pad +1 (8B) -> both staging writes and fragment reads are bank-conflict-free,
    // and each B fragment is ONE aligned ds_load_b64 into a consecutive VGPR pair.
    __shared__ v2f   sB[DIM / 2][NCLS + 1];
    __shared__ float sF[ROWS][DIM + 4];    // +4 pad: b128-aligned staging, conflict-free A reads
    __shared__ float sDiff[NCLS * DIM];    // per-WG class sums
    __shared__ float c2s[NCLS];
    __shared__ float f2s[ROWS];
    __shared__ float red[THREADS];
    __shared__ unsigned cnts[NCLS];

    const int t    = threadIdx.x;
    const int wave = t >> 5;
    const int lane = t & 31;
    const int l16  = lane & 15;
    const int hi   = lane >> 4;

    // Stage centers as paired K: coalesced 8B global reads, conflict-free LDS b64 writes.
    {
        const int kp = t & 63;               // K-pair index
        const int c0 = (t >> 6) * 32;        // two thread-halves split the class range
        for (int c = c0; c < c0 + 32; ++c) {
            v2f v;
            v.x = centers[c * DIM + 2 * kp];
            v.y = centers[c * DIM + 2 * kp + 1];
            sB[kp][c] = v;
        }
    }
    for (int i = t; i < NCLS * DIM; i += THREADS) sDiff[i] = 0.f;
    if (t < NCLS) cnts[t] = 0u;
    __syncthreads();

    if (t < NCLS) {
        float s = 0.f;
        for (int kp = 0; kp < DIM / 2; ++kp) {
            v2f v = sB[kp][t];
            s += v.x * v.x + v.y * v.y;
        }
        c2s[t] = s;
    }
    float lossAcc = 0.f;

    for (int slab = blockIdx.x; slab < SLABS; slab += gridDim.x) {
        const long rowBase = (long)slab * ROWS;
        __syncthreads();                     // protect sF/f2s reuse

        // Prefetch next slab while we work on this one (global_prefetch_b8).
        {
            long nslab = (long)slab + gridDim.x;
            if (nslab < SLABS)
                __builtin_prefetch(&feature[(nslab * ROWS + (t >> 1)) * DIM], 0, 1);
        }

        // Stage feature slab: float4 coalesced loads -> ds_store_b128.
        {
            const int cg = t & 31;           // 32 groups of 4 columns
            for (int r = (t >> 5); r < ROWS; r += 4) {
                const float4 v = *(const float4*)&feature[(rowBase + r) * DIM + cg * 4];
                *(float4*)&sF[r][cg * 4] = v;
            }
        }
        if (t < ROWS) f2s[t] = 0.f;
        __syncthreads();

        // Fused pass: ||f||^2, (center-f) class sums, counts, MSE partials (paired).
        {
            const int r   = t >> 1;
            const int dp0 = (t & 1) * (DIM / 4);     // K-pair range [dp0, dp0+32)
            const int lb  = label[rowBase + r];
            float ssq = 0.f, lp = 0.f;
            for (int dp = dp0; dp < dp0 + DIM / 4; ++dp) {
                const v2f f  = *(const v2f*)&sF[r][2 * dp];
                const v2f cv = sB[dp][lb];
                const float dfx = cv.x - f.x;
                const float dfy = cv.y - f.y;
                ssq += f.x * f.x + f.y * f.y;
                lp  += dfx * dfx + dfy * dfy;
                atomicAdd(&sDiff[lb * DIM + 2 * dp], dfx);      // ds_add_f32
                atomicAdd(&sDiff[lb * DIM + 2 * dp + 1], dfy);
            }
            atomicAdd(&f2s[r], ssq);
            lossAcc += lp;
            if ((t & 1) == 0) atomicAdd(&cnts[lb], 1u);
        }
        __syncthreads();

        // WMMA GEMM: wave w owns rows [w*16, w*16+16), all 64 centers.
        v8f acc[4];
#pragma unroll
        for (int i = 0; i < 4; ++i) acc[i] = (v8f){0.f,0.f,0.f,0.f,0.f,0.f,0.f,0.f};

        const int rw = wave * 16 + l16;      // A-matrix row for this lane
#pragma unroll 4
        for (int k0 = 0; k0 < DIM; k0 += 4) {
            const int kp = (k0 >> 1) + hi;   // lanes 0-15: K=k0,k0+1; 16-31: K=k0+2,k0+3
            v2f a;
            a.x = sF[rw][2 * kp];
            a.y = sF[rw][2 * kp + 1];
#pragma unroll
            for (int ct = 0; ct < 4; ++ct) {
                const v2f b = sB[kp][ct * 16 + l16];   // one ds_load_b64, pre-paired
                acc[ct] = __builtin_amdgcn_wmma_f32_16x16x4_f32(
                    false, a, false, b, (short)0, acc[ct], false, false);
            }
        }

        // Epilogue: distance = f2 + c2 - 2*dot ; coalesced stores.
        float* drow = dist + rowBase * NCLS;
#pragma unroll
        for (int ct = 0; ct < 4; ++ct) {
            const int c     = ct * 16 + l16;
            const float c2v = c2s[c];
#pragma unroll
            for (int j = 0; j < 8; ++j) {
                const int r = wave * 16 + hi * 8 + j;   // D-tile: M = j + 8*hi
                drow[(long)r * NCLS + c] = f2s[r] + c2v - 2.f * acc[ct][j];
            }
        }
    }

    // Flush per-WG accumulators once.
    __syncthreads();
    red[t] = lossAcc;
    __syncthreads();
    if (t == 0) {
        float s = 0.f;
        for (int i = 0; i < THREADS; ++i) s += red[i];
        atomicAdd(gLoss, s);
    }
    if (t < NCLS) atomicAdd(&gCnt[t], cnts[t]);
    for (int i = t; i < NCLS * DIM; i += THREADS)
        atomicAdd(&gDiff[i], sDiff[i]);
}

__global__ void cl2_finalize(const float* __restrict__ gDiff,
                             const unsigned* __restrict__ gCnt,
                             const float* __restrict__ gLoss,
                             float* __restrict__ outLoss,
                             float* __restrict__ outDiff) {
    int i = blockIdx.x * blockDim.x + threadIdx.x;
    if (i < NCLS * DIM) {
        unsigned c = gCnt[i / DIM];
        float cnt = (float)(c < 1u ? 1u : c);
        outDiff[i] = gDiff[i] / cnt;
    }
    if (i == 0) outLoss[0] = gLoss[0] / (float)((long)N_TOK * DIM);
}

extern "C" void kernel_launch(void* const* d_in, const int* in_sizes, int n_in,
                              void* d_out, int out_size, void* d_ws, size_t ws_size,
                              hipStream_t stream) {
    const float* feature = (const float*)d_in[0];
    const int*   label   = (const int*)d_in[1];
    const float* centers = (const float*)d_in[2];

    float* out      = (float*)d_out;
    float* out_loss = out;                      // [1]
    float* out_diff = out + 1;                  // [C*D]
    float* out_dist = out + 1 + NCLS * DIM;     // [N*C]

    float*    gDiff = (float*)d_ws;
    unsigned* gCnt  = (unsigned*)(gDiff + NCLS * DIM);
    float*    gLoss = (float*)(gCnt + NCLS);

    cl2_init_ws<<<(NCLS * DIM + 255) / 256, 256, 0, stream>>>(gDiff, gCnt, gLoss);
    cl2_main<<<NUM_WG, THREADS, 0, stream>>>(feature, label, centers,
                                             out_dist, gDiff, gCnt, gLoss);
    cl2_finalize<<<(NCLS * DIM + 255) / 256, 256, 0, stream>>>(gDiff, gCnt, gLoss,
                                                               out_loss, out_diff);
}